// GraphFilter_81690277970535
// MI455X (gfx1250) — compile-verified
//
#include <hip/hip_runtime.h>

typedef float v2f __attribute__((ext_vector_type(2)));
typedef float v8f __attribute__((ext_vector_type(8)));

#define N_ 8192
#define E_ 256
#define L_ 4

// Kernel 1: s = dot(emb[id], w2), one 256-thread block.
__global__ void gf_dot_s_kernel(const float* __restrict__ emb,
                                const float* __restrict__ w,
                                const int* __restrict__ idp,
                                float* __restrict__ s_out) {
    __shared__ float red[E_];
    const int t = threadIdx.x;
    const int id = *idp;
    red[t] = emb[(size_t)id * E_ + t] * w[E_ + t];   // w2 = w[0, E:]
    __syncthreads();
    for (int off = E_ / 2; off > 0; off >>= 1) {
        if (t < off) red[t] += red[t + off];
        __syncthreads();
    }
    if (t == 0) s_out[0] = red[0];
}

// Kernel 2: r = relu(emb @ w1 + s) via V_WMMA_F32_16X16X4_F32, then
// out[i] = (asum != 0) ? asum * r[i] : 0 with asum = sum_l A[l][id][i].
__global__ void gf_wmma_kernel(const float* __restrict__ emb,
                               const float* __restrict__ A,
                               const int* __restrict__ idp,
                               const float* __restrict__ w,
                               const float* __restrict__ s_in,
                               float* __restrict__ out) {
    __shared__ float w1s[E_];
    const int t = threadIdx.x;
    w1s[t] = w[t];                       // w1 = w[0, :E] staged in LDS
    __syncthreads();

    const int lane  = t & 31;            // wave32
    const int wave  = t >> 5;
    const int sub16 = lane & 15;
    const int half  = lane >> 4;         // 0 or 1
    const int koff  = half << 1;         // K-offset within the 16x4 A tile
    const int row_base = blockIdx.x * 128 + wave * 16;

    // This lane streams row (row_base + sub16) of emb, 2 floats per K-step.
    const float* __restrict__ arow = emb + (size_t)(row_base + sub16) * E_;

    v8f c = {0.f, 0.f, 0.f, 0.f, 0.f, 0.f, 0.f, 0.f};
    #pragma unroll 4
    for (int k = 0; k < E_; k += 4) {
        const int kk = k + koff;
        const v2f a = *(const v2f*)(arow + kk);          // A: 16x4 f32 tile
        const v2f b = *(const v2f*)(&w1s[kk]);           // B: w1 replicated cols
        // (neg_a, A, neg_b, B, c_mod, C, reuse_a, reuse_b)
        c = __builtin_amdgcn_wmma_f32_16x16x4_f32(false, a, false, b,
                                                  (short)0, c, false, false);
    }

    // Every column of D equals dot(row, w1): lane-half h holds rows h*8..h*8+7
    // in c[0..7]. Lanes 0-7 and 16-23 each write one output row.
    const float s  = s_in[0];
    const int   id = *idp;
    if (sub16 < 8) {
        const int m   = half * 8 + sub16;
        const int row = row_base + m;
        const float dotv = c[sub16];                     // dynamic extract (8-way)
        float r = dotv + s;
        r = r > 0.f ? r : 0.f;

        const size_t NN = (size_t)N_ * (size_t)N_;
        const float* __restrict__ ar = A + (size_t)id * N_ + row;
        float asum = 0.f;
        #pragma unroll
        for (int l = 0; l < L_; ++l)
            asum += ar[(size_t)l * NN];

        out[row] = (asum != 0.f) ? asum * r : 0.f;
    }
}

extern "C" void kernel_launch(void* const* d_in, const int* in_sizes, int n_in,
                              void* d_out, int out_size, void* d_ws, size_t ws_size,
                              hipStream_t stream) {
    const float* emb = (const float*)d_in[0];   // (N, E) f32
    const float* A   = (const float*)d_in[1];   // (L, N, N) f32
    const int*   idp = (const int*)d_in[2];     // scalar id
    const float* w   = (const float*)d_in[3];   // (1, 2E) f32
    float* out = (float*)d_out;                 // (1, N) f32
    float* s   = (float*)d_ws;                  // 1 float of scratch

    gf_dot_s_kernel<<<1, E_, 0, stream>>>(emb, w, idp, s);
    gf_wmma_kernel<<<N_ / 128, 256, 0, stream>>>(emb, A, idp, w, s, out);
}